// UnICORNN_12292196401861
// MI455X (gfx1250) — compile-verified
//
#include <hip/hip_runtime.h>
#include <cmath>

// UnICORNN on gfx1250 (MI455X).
// Phase 1: xV = inputs @ V1 via V_WMMA_F32_16X16X4_F32 -> d_ws in [T,B,U] (RT, stays in 192MB L2).
// Phase 2: serial scan, 4 WGs x 8 waves; states + V2 fragments register-resident,
//          xv async-prefetched to LDS (ASYNCcnt), y1 exchanged through LDS,
//          layer-2 GEMM = 4 independent 8-deep f32 WMMA chains, HW v_tanh_f32,
//          NT output stores protect xv's L2 residency.

typedef float v2f __attribute__((ext_vector_type(2)));
typedef float v4f __attribute__((ext_vector_type(4)));
typedef float v8f __attribute__((ext_vector_type(8)));
typedef int   v4i __attribute__((ext_vector_type(4)));

// (neg_a, A, neg_b, B, c_mod, C, reuse_a, reuse_b)
#define WMMA_F32(a, b, c) \
  __builtin_amdgcn_wmma_f32_16x16x4_f32(false, (a), false, (b), (short)0, (c), false, false)

#if defined(__has_builtin)
#if __has_builtin(__builtin_amdgcn_tanhf)
#define TANHF(x) __builtin_amdgcn_tanhf(x)
#elif __has_builtin(__builtin_amdgcn_tanh_f32)
#define TANHF(x) __builtin_amdgcn_tanh_f32(x)
#endif
#if __has_builtin(__builtin_amdgcn_global_load_async_to_lds_b128)
#define HAVE_ASYNC_LDS 1
#endif
#endif
#ifndef TANHF
#define TANHF(x) tanhf(x)
#endif

#if defined(HAVE_ASYNC_LDS)
// builtin signature: (int4 addrspace(1)* src, int4 addrspace(3)* dst, imm offset, imm cpol)
#define GL_ASYNC_B128(gp, lp)                                   \
  __builtin_amdgcn_global_load_async_to_lds_b128(               \
      (__attribute__((address_space(1))) v4i*)(gp),             \
      (__attribute__((address_space(3))) v4i*)(lp), 0, 0)
#if __has_builtin(__builtin_amdgcn_s_wait_asynccnt)
#define WAIT_ASYNC() __builtin_amdgcn_s_wait_asynccnt(0)
#else
#define WAIT_ASYNC() asm volatile("s_wait_asynccnt 0x0" ::: "memory")
#endif
#else
#define WAIT_ASYNC()
#endif

constexpr int B_ = 64, T_ = 2048, F_ = 96, U_ = 128;
constexpr int PAD = 132;  // row stride (floats): conflict-free b64/b128 LDS access, 16B aligned
constexpr float EPS = 0.03f, ALPHA = 0.9f;

__device__ __forceinline__ float mod_tanh_d(float x) {
  return 0.5f + 0.5f * TANHF(0.5f * x);
}

// ---------------- Phase 1: xV[t][b][u] = sum_f inputs[b][t][f] * V1[f][u] ------------
__global__ __launch_bounds__(256) void unicornn_xv_gemm(
    const float* __restrict__ inp,   // [B,T,F]
    const float* __restrict__ V1,    // [F,U]
    float* __restrict__ xv)          // [T,B,U]
{
  const int lane = threadIdx.x & 31;
  const int wv   = threadIdx.x >> 5;
  const int c    = lane & 15;
  const int koff = (lane < 16) ? 0 : 2;   // A/B fragment K sub-offset
  const int mo   = (lane < 16) ? 0 : 8;   // C fragment M sub-offset

  const int  mtile = blockIdx.x * 8 + wv;      // 8192 tiles of 16 rows
  const int  rbase = mtile * 16;
  const long arow  = (long)(rbase + c) * F_;

  // A fragments: 16x4 f32 chunks, K = 96 -> 24 chunks, reused across all 8 N-tiles.
  // Inputs are read exactly once -> NT loads keep them out of L2 (reserved for xv).
  v2f a[24];
#pragma unroll
  for (int kc = 0; kc < 24; ++kc)
    a[kc] = __builtin_nontemporal_load(
        reinterpret_cast<const v2f*>(inp + arow + kc * 4 + koff));

  for (int nt = 0; nt < 8; ++nt) {
    const int n0 = nt * 16;
    v8f acc = {};
#pragma unroll
    for (int kc = 0; kc < 24; ++kc) {
      const int kk = kc * 4 + koff;
      v2f b;
      b.x = V1[kk * U_ + n0 + c];
      b.y = V1[(kk + 1) * U_ + n0 + c];
      acc = WMMA_F32(a[kc], b, acc);
    }
    // Scatter C tile: input row ri = b*T + t  ->  xV row t*B + b  (regular/RT stores:
    // 64 MB xv fits the 192 MB L2 and is re-read once by the scan kernel)
#pragma unroll
    for (int r = 0; r < 8; ++r) {
      const int ri = rbase + r + mo;
      const int bb = ri >> 11;          // / T_
      const int tt = ri & (T_ - 1);
      xv[((long)tt * B_ + bb) * U_ + n0 + c] = acc[r];
    }
  }
}

// Async prefetch of one [16][U_] xv tile (8 KB) into padded LDS rows.
__device__ __forceinline__ void prefetch_xv_tile(const float* __restrict__ xv,
                                                 long tilebase, float (*dst)[PAD],
                                                 int tid) {
#pragma unroll
  for (int i = 0; i < 2; ++i) {
    const int idx = i * 256 + tid;     // 512 x b128 = 8 KB
    const int row = idx >> 5;
    const int seg = idx & 31;
#if defined(HAVE_ASYNC_LDS)
    GL_ASYNC_B128(xv + tilebase + row * U_ + seg * 4, &dst[row][seg * 4]);
#else
    *reinterpret_cast<v4f*>(&dst[row][seg * 4]) =
        *reinterpret_cast<const v4f*>(xv + tilebase + row * U_ + seg * 4);
#endif
  }
}

// ---------------- Phase 2: serial UnICORNN scan over T ------------------------------
__global__ __launch_bounds__(256) void unicornn_scan(
    const float* __restrict__ xv,    // [T,B,U]
    const float* __restrict__ V2,    // [U,U]
    const float* __restrict__ w1, const float* __restrict__ w2,
    const float* __restrict__ b1, const float* __restrict__ b2,
    const float* __restrict__ c1, const float* __restrict__ c2,
    float* __restrict__ out)         // [T,B,4U]  (y1,z1,y2,z2)
{
  __shared__ __align__(16) float xvs[2][16][PAD];  // double-buffered async xv tiles
  __shared__ __align__(16) float y1s[16][PAD];     // y1 exchange for the V2 GEMM

  const int tid  = threadIdx.x;
  const int lane = tid & 31;
  const int wv   = tid >> 5;
  const int c    = lane & 15;
  const int koff = (lane < 16) ? 0 : 2;
  const int mo   = (lane < 16) ? 0 : 8;
  const int n0   = wv * 16;            // this wave's unit-column tile
  const int col  = n0 + c;
  const int b0   = blockIdx.x * 16;    // this WG's batch-row tile

  // per-column constants (broadcast across rows via N=lane in C layout)
  const float w1c = w1[col], w2c = w2[col];
  const float b1c = b1[col], b2c = b2[col];
  const float d1c = EPS * mod_tanh_d(c1[col]);
  const float d2c = EPS * mod_tanh_d(c2[col]);

  // V2 B-fragments, K = 128 -> 32 chunks, resident for the whole scan (64 VGPRs)
  v2f fr2[32];
#pragma unroll
  for (int kc = 0; kc < 32; ++kc) {
    const int kk = kc * 4 + koff;
    fr2[kc].x = V2[kk * U_ + col];
    fr2[kc].y = V2[(kk + 1) * U_ + col];
  }

  v8f y1 = {}, z1 = {}, y2 = {}, z2 = {};

  // prime xv tile for t = 0
  prefetch_xv_tile(xv, (long)b0 * U_, xvs[0], tid);
  WAIT_ASYNC();
  __syncthreads();

  for (int t = 0; t < T_; ++t) {
    const long tb = (long)t * B_;
    const int  p  = t & 1;

    // layer 1 (diagonal recurrence, elementwise in C layout), xv from LDS
#pragma unroll
    for (int r = 0; r < 8; ++r) {
      const float xvv = xvs[p][r + mo][col];
      const float h1  = TANHF(xvv + w1c * y1[r] + b1c);
      z1[r] = z1[r] - d1c * (h1 + ALPHA * y1[r]);
      y1[r] = y1[r] + d1c * z1[r];
    }

    // publish y1 tile for the cross-unit GEMM
#pragma unroll
    for (int r = 0; r < 8; ++r) y1s[r + mo][col] = y1[r];

    // kick async prefetch of next step's xv tile into the other buffer;
    // its latency is hidden behind barrier + GEMM
    const int tn = (t + 1 < T_) ? t + 1 : t;
    prefetch_xv_tile(xv, ((long)tn * B_ + b0) * U_, xvs[p ^ 1], tid);

    __syncthreads();   // y1s ready

    // layer-2 pre-activation: [16,128] @ V2[:,n0:n0+16]
    // 4 independent accumulators -> dependent WMMA chain depth 8 instead of 32
    v8f a0 = {}, a1 = {}, a2 = {}, a3 = {};
#pragma unroll
    for (int kc = 0; kc < 32; kc += 4) {
      v2f fa0 = *reinterpret_cast<const v2f*>(&y1s[c][(kc + 0) * 4 + koff]);
      v2f fa1 = *reinterpret_cast<const v2f*>(&y1s[c][(kc + 1) * 4 + koff]);
      v2f fa2 = *reinterpret_cast<const v2f*>(&y1s[c][(kc + 2) * 4 + koff]);
      v2f fa3 = *reinterpret_cast<const v2f*>(&y1s[c][(kc + 3) * 4 + koff]);
      a0 = WMMA_F32(fa0, fr2[kc + 0], a0);
      a1 = WMMA_F32(fa1, fr2[kc + 1], a1);
      a2 = WMMA_F32(fa2, fr2[kc + 2], a2);
      a3 = WMMA_F32(fa3, fr2[kc + 3], a3);
    }
    const v8f acc = (a0 + a1) + (a2 + a3);

    WAIT_ASYNC();      // my async tile landed
    __syncthreads();   // everyone's async landed; y1s reads done

    // layer 2 elementwise
#pragma unroll
    for (int r = 0; r < 8; ++r) {
      const float h2 = TANHF(acc[r] + w2c * y2[r] + b2c);
      z2[r] = z2[r] - d2c * (h2 + ALPHA * y2[r]);
      y2[r] = y2[r] + d2c * z2[r];
    }

    // emit state [y1|z1|y2|z2]; NT stores: 268 MB write-once stream must not
    // evict the L2-resident xv
#pragma unroll
    for (int r = 0; r < 8; ++r) {
      float* o = out + (tb + b0 + r + mo) * (4 * U_);
      __builtin_nontemporal_store(y1[r], o + col);
      __builtin_nontemporal_store(z1[r], o + U_ + col);
      __builtin_nontemporal_store(y2[r], o + 2 * U_ + col);
      __builtin_nontemporal_store(z2[r], o + 3 * U_ + col);
    }
  }
}

extern "C" void kernel_launch(void* const* d_in, const int* in_sizes, int n_in,
                              void* d_out, int out_size, void* d_ws, size_t ws_size,
                              hipStream_t stream) {
  (void)in_sizes; (void)n_in; (void)out_size; (void)ws_size;
  const float* inp = (const float*)d_in[0];
  const float* w1  = (const float*)d_in[1];
  const float* w2  = (const float*)d_in[2];
  const float* V1  = (const float*)d_in[3];
  const float* V2  = (const float*)d_in[4];
  const float* b1  = (const float*)d_in[5];
  const float* b2  = (const float*)d_in[6];
  const float* c1  = (const float*)d_in[7];
  const float* c2  = (const float*)d_in[8];
  float* out  = (float*)d_out;
  float* xvws = (float*)d_ws;   // needs T_*B_*U_*4 = 64 MiB of scratch

  unicornn_xv_gemm<<<dim3((B_ * T_) / (16 * 8)), dim3(256), 0, stream>>>(inp, V1, xvws);
  unicornn_scan<<<dim3(B_ / 16), dim3(256), 0, stream>>>(xvws, V2, w1, w2, b1, b2, c1, c2, out);
}